// AS19_softmax_62062277427867
// MI455X (gfx1250) — compile-verified
//
#include <hip/hip_runtime.h>

// ---------------------------------------------------------------------------
// 100-term Taylor softmax, N=262144 x C=256 fp32.
// One wave32 per block owns a 16x256 tile:
//   Phase 0: TDM tensor_load_to_lds (global->LDS, HW row padding)
//   Phase 1: WMMA_F32_16X16X4_F32 row-sum reduction (all-ones B) -> mean
//   Phase 2: Horner exp (chain truncated at last nonzero fp32 coefficient --
//            bit-identical to the 100-term reference), e -> LDS
//   Phase 2b: WMMA row-sum of e -> 1/sum
//   Phase 3: coalesced normalize + global_store_b128
// ---------------------------------------------------------------------------

#define N_TERMS 100

struct CoeffArr { float c[N_TERMS]; };
constexpr CoeffArr makeCoeffs() {
  CoeffArr r{};
  double t = 1.0;
  r.c[0] = 1.0f;
  for (int i = 1; i < N_TERMS; ++i) {
    t = t * 0.6931471805599453 / (double)i;   // ln2^i / i!
    r.c[i] = (float)t;
  }
  return r;
}
constexpr CoeffArr kCoef = makeCoeffs();

// Last index whose fp32 coefficient is nonzero (i=34, a denormal; no flush on
// gfx1250). Horner from K_START is bit-identical to starting at c[99]=0:
// the zero prefix computes fma(0, dec, 0) = +0 exactly for all finite dec>=0,
// and NaN dec propagates NaN through both variants.
constexpr int lastNonZeroCoeff() {
  int k = 0;
  for (int i = 0; i < N_TERMS; ++i)
    if (kCoef.c[i] != 0.0f) k = i;
  return k;
}
constexpr int K_START = lastNonZeroCoeff();

typedef __attribute__((ext_vector_type(2))) float v2f;
typedef __attribute__((ext_vector_type(8))) float v8f;
typedef unsigned int v4u __attribute__((ext_vector_type(4)));
typedef int          v8i __attribute__((ext_vector_type(8)));
typedef int          v4i __attribute__((ext_vector_type(4)));

static constexpr int   PITCHF = 260;  // 256 + 4 floats pad per row (TDM pad_amount=3)
static constexpr float kLog2e = 1.4426950408889634f;

#if defined(__has_builtin)
#if __has_builtin(__builtin_amdgcn_tensor_load_to_lds) && __has_builtin(__builtin_amdgcn_s_wait_tensorcnt)
#define HAVE_TDM 1
#endif
#endif
#ifndef HAVE_TDM
#define HAVE_TDM 0
#endif

// Extract lane's needed row value from WMMA D (VGPR r holds rows r / r+8), q = lane&7.
__device__ __forceinline__ float pick8(v8f d, int q) {
  float p0 = (q & 1) ? d[1] : d[0];
  float p1 = (q & 1) ? d[3] : d[2];
  float p2 = (q & 1) ? d[5] : d[4];
  float p3 = (q & 1) ? d[7] : d[6];
  float r0 = (q & 2) ? p1 : p0;
  float r1 = (q & 2) ? p3 : p2;
  return (q & 4) ? r1 : r0;
}

// Same math as reference: a = log2e*|xc|, dec = frac(a), Horner in dec with the
// reference's fp32-cast coefficients, scale by 2^trunc(a), reciprocal if xc < 0.
__device__ __forceinline__ float softexp(float xc) {
  float aa  = kLog2e * fabsf(xc);
  float tr  = truncf(aa);
  float dec = aa - tr;
  float ii  = fminf(fmaxf(tr, -150.0f), 125.0f);
  float p = kCoef.c[K_START];
#pragma unroll
  for (int i = K_START - 1; i >= 0; --i)
    p = __builtin_fmaf(p, dec, kCoef.c[i]);     // v_pk_fma_f32 pairs, literal coeffs
  float e = ldexpf(p, (int)ii);                 // exact 2^int scale
  return (xc < 0.0f) ? __builtin_amdgcn_rcpf(e) : e;
}

__global__ __launch_bounds__(32) void softmax100_wmma(const float* __restrict__ x,
                                                      float* __restrict__ out) {
  __shared__ float T[16 * PITCHF];              // 16,640 B, one tile per wave/block
  const int l = threadIdx.x;                    // 0..31
  const int tile = blockIdx.x;

  const int q    = l & 7;
  const int hi   = (l >> 4) & 1;                // A layout: high half = K2,K3
  const int b3   = (l >> 3) & 1;                // D layout: column half
  const int rowA = l & 15;                      // A layout row (fixed by HW)
  const int rowD = q + 8 * hi;                  // D layout row this lane owns
  const int permSrc = (b3 ^ hi) ? (l ^ 24) : l; // D partials -> A-layout lanes

  const float4* gx = (const float4*)(x   + (size_t)tile * 4096);
  float4*       gy = (float4*)(out + (size_t)tile * 4096);

  // ---- Phase 0: tile load global -> LDS via Tensor Data Mover -----------
#if HAVE_TDM
  {
    unsigned long long ga = (unsigned long long)(x + (size_t)tile * 4096);
    unsigned int ldsOff = (unsigned int)(unsigned long long)(void*)T; // AS3 byte offset

    v4u g0;
    g0[0] = 1u;                                            // count=1, user mode
    g0[1] = ldsOff;                                        // lds_addr
    g0[2] = (unsigned int)(ga & 0xFFFFFFFFu);              // global_addr[31:0]
    g0[3] = (unsigned int)((ga >> 32) & 0x01FFFFFFu)       // global_addr[56:32]
            | (2u << 30);                                  // type = 2 ("image")

    v8i g1;
    g1[0] = (int)((2u << 16)      // data_size = 2 -> 4-byte elements
                | (1u << 20)      // pad_enable
                | (7u << 22)      // pad_interval: every 256 DWORDs (one row)
                | (3u << 25));    // pad_amount: 4 DWORDs  -> PITCHF = 260
    g1[1] = (int)(256u << 16);    // tensor_dim0 = 256 (bits 79:48, low half)
    g1[2] = (int)(16u  << 16);    // tensor_dim0 hi = 0 | tensor_dim1 = 16
    g1[3] = (int)(256u << 16);    // tensor_dim1 hi = 0 | tile_dim0 = 256
    g1[4] = 16;                   // tile_dim1 = 16, tile_dim2 = 0
    g1[5] = 256;                  // tensor_dim0_stride[31:0] = 256
    g1[6] = 0;                    // stride0 hi | tensor_dim1_stride lo
    g1[7] = 0;

    v4i gz4 = {0, 0, 0, 0};       // groups 2/3 unused (2D tile)
    v8i gz8 = {0, 0, 0, 0, 0, 0, 0, 0};
    __builtin_amdgcn_tensor_load_to_lds(g0, g1, gz4, gz4, gz8, 0);
    __builtin_amdgcn_s_wait_tensorcnt(0);
  }
#else
#pragma unroll
  for (int i = 0; i < 32; ++i) {
    float4 v = gx[i * 32 + l];                  // 512B contiguous per instruction
    int row = i >> 1;
    int col = (i & 1) * 128 + l * 4;
    *(float4*)&T[row * PITCHF + col] = v;
  }
#endif
  __syncthreads();                              // single-wave: compiler ordering fence

  // ---- Phase 1: partial row sums of x in A-matrix layout, WMMA reduce ---
  float s0 = 0.0f, s1 = 0.0f;
  const float* aBase = &T[rowA * PITCHF + hi * 128];
#pragma unroll
  for (int i = 0; i < 16; ++i) {
    float4 v = *(const float4*)&aBase[i * 4];
    s0 += (v.x + v.y) + (v.z + v.w);            // K-slot 2*hi
  }
#pragma unroll
  for (int i = 16; i < 32; ++i) {
    float4 v = *(const float4*)&aBase[i * 4];
    s1 += (v.x + v.y) + (v.z + v.w);            // K-slot 2*hi+1
  }

  v2f ones; ones[0] = 1.0f; ones[1] = 1.0f;     // all-ones B: D[m][n] = sum_k A[m][k]
  v2f av;   av[0]   = s0;   av[1]   = s1;
  v8f cz = {};
  v8f dsum = __builtin_amdgcn_wmma_f32_16x16x4_f32(false, av, false, ones,
                                                   (short)0, cz, false, false);
  const float mean = pick8(dsum, q) * (1.0f / 256.0f);
  __syncthreads();

  // ---- Phase 2: Horner exp; overwrite LDS with e; partial e-sums --------
  // Prefetch next float4 one iteration ahead (row pad makes overread in-bounds).
  float e0 = 0.0f, e1 = 0.0f;
  float* dBase = &T[rowD * PITCHF + b3 * 128];
  float4 cur = *(const float4*)&dBase[0];
#pragma unroll 1
  for (int i = 0; i < 16; ++i) {
    float4 v = cur;
    cur = *(const float4*)&dBase[(i + 1) * 4];
    float4 e;
    e.x = softexp(v.x - mean);
    e.y = softexp(v.y - mean);
    e.z = softexp(v.z - mean);
    e.w = softexp(v.w - mean);
    *(float4*)&dBase[i * 4] = e;
    e0 += (e.x + e.y) + (e.z + e.w);
  }
#pragma unroll 1
  for (int i = 16; i < 32; ++i) {
    float4 v = cur;
    cur = *(const float4*)&dBase[(i + 1) * 4];  // i=31 overreads into row pad: safe
    float4 e;
    e.x = softexp(v.x - mean);
    e.y = softexp(v.y - mean);
    e.z = softexp(v.z - mean);
    e.w = softexp(v.w - mean);
    *(float4*)&dBase[i * 4] = e;
    e1 += (e.x + e.y) + (e.z + e.w);
  }

  // Permute e-partials from D-assignment into A-matrix layout, WMMA reduce.
  float ae0 = __shfl(e0, permSrc, 32);
  float ae1 = __shfl(e1, permSrc, 32);
  v2f aev; aev[0] = ae0; aev[1] = ae1;
  v8f dsum2 = __builtin_amdgcn_wmma_f32_16x16x4_f32(false, aev, false, ones,
                                                    (short)0, cz, false, false);
  const float inv = __builtin_amdgcn_rcpf(pick8(dsum2, q));
  __syncthreads();

  // ---- Phase 3: normalize + coalesced store -----------------------------
#pragma unroll
  for (int i = 0; i < 32; ++i) {
    int row = i >> 1;
    float invr = __shfl(inv, (row & 7) + ((row >> 3) << 4), 32); // v_readlane
    int col = (i & 1) * 128 + l * 4;
    float4 v = *(const float4*)&T[row * PITCHF + col];
    v.x *= invr; v.y *= invr; v.z *= invr; v.w *= invr;
    gy[i * 32 + l] = v;                          // 512B contiguous per instruction
  }
}

extern "C" void kernel_launch(void* const* d_in, const int* in_sizes, int n_in,
                              void* d_out, int out_size, void* d_ws, size_t ws_size,
                              hipStream_t stream) {
  (void)n_in; (void)out_size; (void)d_ws; (void)ws_size;
  const float* x = (const float*)d_in[0];
  float* out = (float*)d_out;
  const int rows   = in_sizes[0] / 256;          // C = 256
  const int nTiles = rows / 16;                  // one 16x256 tile per wave/block
  softmax100_wmma<<<nTiles, 32, 0, stream>>>(x, out);
}